// ILN_1580547973998
// MI455X (gfx1250) — compile-verified
//
#include <hip/hip_runtime.h>

#define EPS 1e-8f

typedef float v2f __attribute__((ext_vector_type(2)));
typedef float v8f __attribute__((ext_vector_type(8)));
typedef float f4  __attribute__((ext_vector_type(4)));

constexpr int CCH      = 256;     // channels
constexpr int HW       = 64 * 64; // 4096 spatial elements per (n,c)
constexpr int NTHREADS = 256;     // 8 waves (wave32)

// ---------------------------------------------------------------------------
// Pass 1: per-(n,c) sum and sum-of-squares.
// One block per instance; each thread reduces 16 floats (4x float4), then the
// wave reduces its 32 (s,q) pairs with a single V_WMMA_F32_16X16X4_F32:
//   A[m][0..3] = { s_m, q_m, s_{m+16}, q_{m+16} }   (32-bit A 16x4 layout)
//   B picked so D[:,0] = s-halves summed, D[:,1] = q-halves summed
//     -> B nonzero at (K0,N0),(K2,N0),(K1,N1),(K3,N1) == lanes {0,17} in both
//        B VGPRs under the 32-bit B layout.
// Column totals finish with 8 accumulator adds + one shfl_xor(16).
// ---------------------------------------------------------------------------
__global__ void iln_stats_kernel(const float* __restrict__ x,
                                 float* __restrict__ i_sum,
                                 float* __restrict__ i_sumsq) {
  const int inst = blockIdx.x;                   // n*C + c
  const int t    = threadIdx.x;
  const f4* xp4  = (const f4*)(x + (size_t)inst * HW);

  float s = 0.f, q = 0.f;
#pragma unroll
  for (int k = 0; k < 4; ++k) {
    f4 v = xp4[k * NTHREADS + t];
    s += v.x + v.y + v.z + v.w;
    q += v.x * v.x + v.y * v.y + v.z * v.z + v.w * v.w;
  }

  const int lane = t & 31;
  const int wid  = t >> 5;

  v2f a;
  a.x = s;
  a.y = q;
  const float bsel = (lane == 0 || lane == 17) ? 1.0f : 0.0f;
  v2f b;
  b.x = bsel;
  b.y = bsel;
  v8f c = {};
  // 8 args: (neg_a, A, neg_b, B, c_mod, C, reuse_a, reuse_b)
  v8f d = __builtin_amdgcn_wmma_f32_16x16x4_f32(false, a, false, b,
                                                (short)0, c, false, false);
  float r = 0.f;
#pragma unroll
  for (int j = 0; j < 8; ++j) r += d[j];
  // lanes 0/16 hold the two 8-row halves of the s column; lanes 1/17 of q.
  r += __shfl_xor(r, 16, 32);
  const float S = __shfl(r, 0, 32);   // wave total sum
  const float Q = __shfl(r, 1, 32);   // wave total sumsq

  __shared__ float sh_s[NTHREADS / 32];
  __shared__ float sh_q[NTHREADS / 32];
  if (lane == 0) {
    sh_s[wid] = S;
    sh_q[wid] = Q;
  }
  __syncthreads();
  if (t == 0) {
    float ts = 0.f, tq = 0.f;
#pragma unroll
    for (int w = 0; w < NTHREADS / 32; ++w) {
      ts += sh_s[w];
      tq += sh_q[w];
    }
    i_sum[inst]   = ts;
    i_sumsq[inst] = tq;
  }
}

// ---------------------------------------------------------------------------
// Pass 2: fold instance stats into layer stats per n, emit per-(n,c)
// scale/bias so the apply pass is a single FMA. 32 blocks x 256 threads.
// ---------------------------------------------------------------------------
__global__ void iln_coeff_kernel(const float* __restrict__ i_sum,
                                 const float* __restrict__ i_sumsq,
                                 const float* __restrict__ rho,
                                 const float* __restrict__ gamma,
                                 const float* __restrict__ beta,
                                 float* __restrict__ scale,
                                 float* __restrict__ bias) {
  const int n   = blockIdx.x;
  const int c   = threadIdx.x;
  const int idx = n * CCH + c;

  const float s = i_sum[idx];
  const float q = i_sumsq[idx];

  __shared__ float ls[CCH];
  __shared__ float lq[CCH];
  ls[c] = s;
  lq[c] = q;
  __syncthreads();
  for (int off = CCH / 2; off > 0; off >>= 1) {
    if (c < off) {
      ls[c] += ls[c + off];
      lq[c] += lq[c + off];
    }
    __syncthreads();
  }

  const float inv_hw  = 1.0f / (float)HW;
  const float inv_chw = 1.0f / (float)(CCH * HW);

  const float mi  = s * inv_hw;
  const float vi  = fmaxf(q * inv_hw - mi * mi, 0.f);
  const float rsi = rsqrtf(vi + EPS);

  const float ml  = ls[0] * inv_chw;
  const float vl  = fmaxf(lq[0] * inv_chw - ml * ml, 0.f);
  const float rsl = rsqrtf(vl + EPS);

  const float r  = rho[c];
  const float g  = gamma[c];
  const float bt = beta[c];

  scale[idx] = g * (r * rsi + (1.f - r) * rsl);
  bias[idx]  = bt - g * (r * mi * rsi + (1.f - r) * ml * rsl);
}

// ---------------------------------------------------------------------------
// Pass 3: out = fma(x, scale, bias). Non-temporal loads (last use of x) and
// non-temporal stores (out never re-read) keep x resident in the 192 MB L2
// while the pass streams, minimizing HBM traffic.
// ---------------------------------------------------------------------------
__global__ void iln_apply_kernel(const float* __restrict__ x,
                                 const float* __restrict__ scale,
                                 const float* __restrict__ bias,
                                 float* __restrict__ out) {
  const int inst = blockIdx.x;
  const float sc = scale[inst];
  const float bi = bias[inst];

  const f4* xp = (const f4*)(x + (size_t)inst * HW);
  f4* op       = (f4*)(out + (size_t)inst * HW);
  const int t  = threadIdx.x;

#pragma unroll
  for (int k = 0; k < 4; ++k) {
    f4 v = __builtin_nontemporal_load(&xp[k * NTHREADS + t]);
    f4 o;
    o.x = fmaf(v.x, sc, bi);
    o.y = fmaf(v.y, sc, bi);
    o.z = fmaf(v.z, sc, bi);
    o.w = fmaf(v.w, sc, bi);
    __builtin_nontemporal_store(o, &op[k * NTHREADS + t]);
  }
}

extern "C" void kernel_launch(void* const* d_in, const int* in_sizes, int n_in,
                              void* d_out, int out_size, void* d_ws, size_t ws_size,
                              hipStream_t stream) {
  const float* x     = (const float*)d_in[0];
  const float* rho   = (const float*)d_in[1];
  const float* gamma = (const float*)d_in[2];
  const float* beta  = (const float*)d_in[3];
  float* out = (float*)d_out;
  float* ws  = (float*)d_ws;

  const int N     = in_sizes[0] / (CCH * HW); // 32
  const int insts = N * CCH;                  // 8192

  float* i_sum   = ws;
  float* i_sumsq = ws + insts;
  float* scale   = ws + 2 * insts;
  float* bias    = ws + 3 * insts;

  iln_stats_kernel<<<insts, NTHREADS, 0, stream>>>(x, i_sum, i_sumsq);
  iln_coeff_kernel<<<N, CCH, 0, stream>>>(i_sum, i_sumsq, rho, gamma, beta,
                                          scale, bias);
  iln_apply_kernel<<<insts, NTHREADS, 0, stream>>>(x, scale, bias, out);
}